// DistMultScore_52630529245700
// MI455X (gfx1250) — compile-verified
//
#include <hip/hip_runtime.h>

// DistMult edge scoring for MI455X (gfx1250).
//
// Roofline: ~6 KB read + 4 B written per edge vs ~1.5 KFLOP per edge
// (0.25 FLOP/byte) -> memory bound by ~100x at 23.3 TB/s. No WMMA: the only
// matmul mapping (diagonal of 16x16) wastes 15/16 of MACs and halves load
// width. Optimization targets are pure data movement:
//   * node_emb (205 MB) ~fits the 192 MB L2 and has ~3x gather reuse ->
//     default (RT) cache policy.
//   * rel_emb (307 MB) is a single-use stream -> th:TH_LOAD_NT so it does
//     not evict node rows from L2.
//   * CDNA5 async tensor path: GLOBAL_LOAD_ASYNC_TO_LDS_B128 double-buffered
//     per wave, retired with S_WAIT_ASYNCCNT, to software-pipeline the
//     latency-dominated gathers.

#define HIDDEN           512
#define ROW_BYTES        (HIDDEN * 4)        // 2048 B per embedding row
#define WAVES_PER_BLOCK  4
#define BLOCK_THREADS    (WAVES_PER_BLOCK * 32)
#define STAGE_FLOATS     (3 * HIDDEN)        // head, rel, tail rows
#define EDGES_PER_WAVE   8                   // target depth of grid-stride loop

// Low 32 bits of a generic pointer to __shared__ data are the LDS byte offset
// (LDS aperture lives in addr[63:32] on gfx1250).
__device__ __forceinline__ unsigned lds_off(const void* p) {
    return (unsigned)(unsigned long long)p;
}

// Copy one 2048 B row (global -> LDS) with 4 async b128 ops per lane group:
// lane l moves bytes [l*16 + i*512], i = 0..3  (32 lanes x 16 B = 512 B/op).
__device__ __forceinline__ void async_row_rt(unsigned ldsDst, const float* g,
                                             unsigned laneByte) {
#pragma unroll
    for (int i = 0; i < 4; ++i) {
        unsigned o = laneByte + (unsigned)i * 512u;
        asm volatile("global_load_async_to_lds_b128 %0, %1, off"
                     :
                     : "v"(ldsDst + o), "v"((const char*)g + o)
                     : "memory");
    }
}

__device__ __forceinline__ void async_row_nt(unsigned ldsDst, const float* g,
                                             unsigned laneByte) {
#pragma unroll
    for (int i = 0; i < 4; ++i) {
        unsigned o = laneByte + (unsigned)i * 512u;
        asm volatile("global_load_async_to_lds_b128 %0, %1, off th:TH_LOAD_NT"
                     :
                     : "v"(ldsDst + o), "v"((const char*)g + o)
                     : "memory");
    }
}

__device__ __forceinline__ void prefetch_edge(unsigned ldsBase,
                                              const float* __restrict__ node_emb,
                                              const float* __restrict__ rel_emb,
                                              const long long* __restrict__ src,
                                              const long long* __restrict__ dst,
                                              int e, unsigned laneByte) {
    long long s = src[e];
    long long d = dst[e];
    async_row_rt(ldsBase,                 node_emb + s * (long long)HIDDEN, laneByte);
    async_row_nt(ldsBase + ROW_BYTES,     rel_emb  + (long long)e * HIDDEN, laneByte);
    async_row_rt(ldsBase + 2 * ROW_BYTES, node_emb + d * (long long)HIDDEN, laneByte);
}

__global__ void __launch_bounds__(BLOCK_THREADS)
distmult_score_kernel(const float* __restrict__ node_emb,
                      const float* __restrict__ rel_emb,
                      const long long* __restrict__ src,
                      const long long* __restrict__ dst,
                      float* __restrict__ out,
                      int nEdges, int totalWaves) {
    // Per-wave double-buffered staging: 2 stages x 3 rows x 2 KB = 12 KB/wave,
    // 48 KB/block (well under the 320 KB WGP LDS pool).
    __shared__ __align__(16) float smem[WAVES_PER_BLOCK][2][STAGE_FLOATS];

    const int lane = (int)(threadIdx.x & 31u);
    const int wib  = (int)(threadIdx.x >> 5);              // wave in block
    const int wid  = (int)blockIdx.x * WAVES_PER_BLOCK + wib;
    const unsigned laneByte = (unsigned)lane * 16u;

    int e = wid;                    // wave-uniform -> no divergence anywhere
    if (e >= nEdges) return;

    const unsigned base[2] = { lds_off(&smem[wib][0][0]),
                               lds_off(&smem[wib][1][0]) };

    prefetch_edge(base[0], node_emb, rel_emb, src, dst, e, laneByte);

    int cur = 0;
    while (true) {
        const int  en   = e + totalWaves;
        const bool more = en < nEdges;

        if (more) {
            // Launch next stage, then wait only for the current stage's 12 ops
            // (the newer 12 remain in flight behind the compute below).
            prefetch_edge(base[cur ^ 1], node_emb, rel_emb, src, dst, en, laneByte);
            asm volatile("s_wait_asynccnt 12" ::: "memory");
        } else {
            asm volatile("s_wait_asynccnt 0" ::: "memory");
        }

        // Compute from the completed stage: lane l handles 16 of 512 dims.
        const float*  buf = &smem[wib][cur][0];
        const float4* h4  = (const float4*)(buf);
        const float4* r4  = (const float4*)(buf + HIDDEN);
        const float4* t4  = (const float4*)(buf + 2 * HIDDEN);

        float acc = 0.0f;
#pragma unroll
        for (int i = 0; i < 4; ++i) {
            const int idx = lane + 32 * i;
            float4 h = h4[idx];
            float4 r = r4[idx];
            float4 t = t4[idx];
            acc = fmaf(h.x * r.x, t.x, acc);
            acc = fmaf(h.y * r.y, t.y, acc);
            acc = fmaf(h.z * r.z, t.z, acc);
            acc = fmaf(h.w * r.w, t.w, acc);
        }

        // wave32 butterfly reduction (5 steps).
#pragma unroll
        for (int o = 16; o > 0; o >>= 1)
            acc += __shfl_xor(acc, o, 32);

        if (lane == 0)
            __builtin_nontemporal_store(acc, &out[e]);   // 600 KB stream, no reuse

        if (!more) break;
        e   = en;
        cur ^= 1;
    }
}

extern "C" void kernel_launch(void* const* d_in, const int* in_sizes, int n_in,
                              void* d_out, int out_size, void* d_ws, size_t ws_size,
                              hipStream_t stream) {
    const float*     node_emb = (const float*)d_in[0];      // [N_NODES, 512] f32
    const float*     rel_emb  = (const float*)d_in[1];      // [N_EDGES, 512] f32
    const long long* src      = (const long long*)d_in[2];  // [N_EDGES] i64
    const long long* dst      = (const long long*)d_in[3];  // [N_EDGES] i64
    float*           out      = (float*)d_out;              // [N_EDGES] f32

    const int E = in_sizes[2];

    int wavesWanted = (E + EDGES_PER_WAVE - 1) / EDGES_PER_WAVE;
    int blocks      = (wavesWanted + WAVES_PER_BLOCK - 1) / WAVES_PER_BLOCK;
    if (blocks < 1) blocks = 1;
    const int totalWaves = blocks * WAVES_PER_BLOCK;

    distmult_score_kernel<<<blocks, BLOCK_THREADS, 0, stream>>>(
        node_emb, rel_emb, src, dst, out, E, totalWaves);
}